// VariationalGraohAutoEncoder_47141561041366
// MI455X (gfx1250) — compile-verified
//
#include <hip/hip_runtime.h>
#include <math.h>

typedef float v2f __attribute__((ext_vector_type(2)));
typedef float v8f __attribute__((ext_vector_type(8)));

static constexpr int   NN = 100000;   // nodes
static constexpr int   NE = 1600000;  // edges
static constexpr float NEG = 0.2f;    // leaky-relu slope

__device__ __forceinline__ float leaky(float x) { return x > 0.f ? x : NEG * x; }
// order-preserving float<->uint map for atomicMax on floats of any sign
__device__ __forceinline__ unsigned f2ord(float f) {
  unsigned u = __float_as_uint(f);
  return (u & 0x80000000u) ? ~u : (u | 0x80000000u);
}
__device__ __forceinline__ float ord2f(unsigned v) {
  unsigned u = (v & 0x80000000u) ? (v & 0x7fffffffu) : ~v;
  return __uint_as_float(u);
}

// ---------------- elementwise helpers ----------------
__global__ void zero_f32_k(float* p, int n) {
  int i = blockIdx.x * blockDim.x + threadIdx.x;
  if (i < n) p[i] = 0.f;
}
__global__ void relu_k(float* p, int n) {
  int i = blockIdx.x * blockDim.x + threadIdx.x;
  if (i < n) p[i] = fmaxf(p[i], 0.f);
}
// r = relu(h2) + r + br[c]   (r initially holds x@Wr)
__global__ void residual_k(const float* __restrict__ h2, float* __restrict__ r,
                           const float* __restrict__ br, int n) {
  int i = blockIdx.x * blockDim.x + threadIdx.x;
  if (i < n) { int c = i & 127; r[i] = fmaxf(h2[i], 0.f) + r[i] + br[c]; }
}

// ---------------- shared edge_attr mean (PyG fill_value='mean') ----------------
__global__ void attr_accum_k(const int* __restrict__ dst, const float* __restrict__ ea,
                             float* attr_sum, float* cnt) {
  int i = blockIdx.x * blockDim.x + threadIdx.x;
  if (i >= NE * 16) return;
  int e = i >> 4, j = i & 15;
  int d = dst[e];
  atomicAdd(&attr_sum[(size_t)d * 16 + j], ea[i]);
  if (j == 0) atomicAdd(&cnt[d], 1.f);
}
__global__ void attr_mean_k(float* attr_sum, const float* cnt) {
  int i = blockIdx.x * blockDim.x + threadIdx.x;
  if (i < NN * 16) attr_sum[i] /= fmaxf(cnt[i >> 4], 1.f);
}

// wa[i] = sum_j We[i,j] * ae[j]   (collapses (ea@We)@ae -> ea@wa)
__global__ void wa_k(const float* __restrict__ We, const float* __restrict__ ae,
                     float* wa, int K) {
  int i = threadIdx.x;
  if (i < 16) { float s = 0.f; for (int j = 0; j < K; ++j) s += We[i * K + j] * ae[j]; wa[i] = s; }
}

// ---------------- WMMA f32 GEMM: C[N,COUT] = A[N,128] @ B[128,COUT] ----------------
template <int COLT>  // COLT = COUT/16; blockDim = 32*COLT; one 16-row stripe per block
__global__ void gemm_k128(const float* __restrict__ A, const float* __restrict__ B,
                          float* __restrict__ C) {
  constexpr int COUT = COLT * 16;
  __shared__ float As[16 * 128];
  const int tid = threadIdx.x;
  const size_t abase = (size_t)blockIdx.x * (16 * 128);
  for (int i = tid * 4; i < 16 * 128; i += blockDim.x * 4)
    *(float4*)(&As[i]) = *(const float4*)(&A[abase + i]);
  __syncthreads();

  const int wave = tid >> 5, lane = tid & 31, half = lane >> 4, lidx = lane & 15;
  const int n0 = wave * 16;
  v8f acc = {};
#pragma unroll
  for (int k0 = 0; k0 < 128; k0 += 4) {
    v2f a, b;
    // A 16x4 f32 frag: lanes 0-15 -> K=k0,k0+1 ; lanes 16-31 -> K=k0+2,k0+3 ; M = lidx
    a.x = As[lidx * 128 + k0 + 2 * half];
    a.y = As[lidx * 128 + k0 + 2 * half + 1];
    // B 4x16 f32 frag: rows striped across lanes; N = n0+lidx
    b.x = B[(k0 + 2 * half) * COUT + n0 + lidx];
    b.y = B[(k0 + 2 * half + 1) * COUT + n0 + lidx];
    acc = __builtin_amdgcn_wmma_f32_16x16x4_f32(false, a, false, b, (short)0, acc,
                                                false, false);
  }
  const size_t row0 = (size_t)blockIdx.x * 16;
#pragma unroll
  for (int r = 0; r < 8; ++r)
    C[(row0 + r + 8 * half) * COUT + n0 + lidx] = acc[r];
}

// ---------------- per-node prep: attention scalars + self-loop logit ----------------
template <int C, bool ET>
__global__ void node_prep_k(const float* __restrict__ h, const float* __restrict__ a_src,
                            const float* __restrict__ a_dst,
                            const float* __restrict__ mean_attr,
                            const float* __restrict__ wa, float* s_src, float* s_dst,
                            float* self_logit, unsigned* segmax_u) {
  int gid = blockIdx.x * blockDim.x + threadIdx.x;
  int n = gid >> 5, lane = gid & 31;
  if (n >= NN) return;
  float ps = 0.f, pd = 0.f;
#pragma unroll
  for (int i = 0; i < C / 32; ++i) {
    int c = lane + i * 32;
    float v = h[(size_t)n * C + c];
    ps += v * a_src[c];
    pd += v * a_dst[c];
  }
#pragma unroll
  for (int m = 16; m > 0; m >>= 1) { ps += __shfl_xor(ps, m, 32); pd += __shfl_xor(pd, m, 32); }
  if (lane == 0) {
    float st = 0.f;
    if (ET) { for (int j = 0; j < 16; ++j) st += mean_attr[(size_t)n * 16 + j] * wa[j]; }
    float sl = leaky(ps + pd + st);
    s_src[n] = ps; s_dst[n] = pd; self_logit[n] = sl; segmax_u[n] = f2ord(sl);
  }
}

// eterm[e] = dot(edge_attr[e,:16], wa)  -- 16 consecutive lanes per edge, fully coalesced
__global__ void eterm_k(const float* __restrict__ ea, const float* __restrict__ wa,
                        float* eterm) {
  int i = blockIdx.x * blockDim.x + threadIdx.x;
  if (i >= NE * 16) return;
  int j = i & 15;
  float v = ea[i] * wa[j];
  v += __shfl_xor(v, 1, 32); v += __shfl_xor(v, 2, 32);
  v += __shfl_xor(v, 4, 32); v += __shfl_xor(v, 8, 32);
  if (j == 0) eterm[i >> 4] = v;
}

template <bool ET>
__global__ void edge_max_k(const int* __restrict__ src, const int* __restrict__ dst,
                           const float* __restrict__ s_src, const float* __restrict__ s_dst,
                           const float* __restrict__ eterm, unsigned* segmax_u) {
  int e = blockIdx.x * blockDim.x + threadIdx.x;
  if (e >= NE) return;
  int d = dst[e];
  float l = s_src[src[e]] + s_dst[d];
  if (ET) l += eterm[e];
  atomicMax(&segmax_u[d], f2ord(leaky(l)));
}

__global__ void node_mid_k(const unsigned* __restrict__ segmax_u,
                           const float* __restrict__ self_logit, float* segmaxf,
                           float* segsum) {
  int n = blockIdx.x * blockDim.x + threadIdx.x;
  if (n >= NN) return;
  float m = ord2f(segmax_u[n]);
  segmaxf[n] = m;
  segsum[n] = __expf(self_logit[n] - m);  // self-loop exp term
}

template <bool ET>
__global__ void edge_sum_k(const int* __restrict__ src, const int* __restrict__ dst,
                           const float* __restrict__ s_src, const float* __restrict__ s_dst,
                           const float* __restrict__ eterm,
                           const float* __restrict__ segmaxf, float* segsum) {
  int e = blockIdx.x * blockDim.x + threadIdx.x;
  if (e >= NE) return;
  int d = dst[e];
  float l = s_src[src[e]] + s_dst[d];
  if (ET) l += eterm[e];
  atomicAdd(&segsum[d], __expf(leaky(l) - segmaxf[d]));
}

// out[n,:] = alpha_self * h[n,:] + bias  (scatter pass then adds edge contributions)
template <int C>
__global__ void out_init_k(const float* __restrict__ h, const float* __restrict__ self_logit,
                           const float* __restrict__ segmaxf, const float* __restrict__ segsum,
                           const float* __restrict__ bias, float* __restrict__ out) {
  int i = blockIdx.x * blockDim.x + threadIdx.x;
  if (i >= NN * (C / 4)) return;
  int n = i / (C / 4), q = i % (C / 4);
  float a = __expf(self_logit[n] - segmaxf[n]) / (segsum[n] + 1e-16f);
  float4 hv = *(const float4*)&h[(size_t)n * C + q * 4];
  float4 o;
  o.x = a * hv.x + bias[q * 4 + 0];
  o.y = a * hv.y + bias[q * 4 + 1];
  o.z = a * hv.z + bias[q * 4 + 2];
  o.w = a * hv.w + bias[q * 4 + 3];
  *(float4*)&out[(size_t)n * C + q * 4] = o;
}

// wave-per-edge weighted scatter-add; lanes own consecutive channels (coalesced atomics)
template <int C, bool ET>
__global__ void edge_agg_k(const int* __restrict__ src, const int* __restrict__ dst,
                           const float* __restrict__ s_src, const float* __restrict__ s_dst,
                           const float* __restrict__ eterm,
                           const float* __restrict__ segmaxf, const float* __restrict__ segsum,
                           const float* __restrict__ h, float* __restrict__ out) {
  int gid = blockIdx.x * blockDim.x + threadIdx.x;
  int e = gid >> 5, lane = gid & 31;
  if (e >= NE) return;
  int s = src[e], d = dst[e];
  float l = s_src[s] + s_dst[d];
  if (ET) l += eterm[e];
  float alpha = __expf(leaky(l) - segmaxf[d]) / (segsum[d] + 1e-16f);
#pragma unroll
  for (int i = 0; i < C / 32; ++i) {
    int c = lane + i * 32;
    atomicAdd(&out[(size_t)d * C + c], alpha * h[(size_t)s * C + c]);
  }
}

// ---------------- host-side conv driver ----------------
struct Scratch {
  float *s_src, *s_dst, *self_l, *segmaxf, *segsum;
  unsigned* segmax_u;
  const float* mean_attr;
  float *eterm, *wa;
};

template <int C, bool ET>
static void run_gat(hipStream_t st, const float* hin, const float* Wm, const float* asrc,
                    const float* adst, const float* bias, const float* We, const float* ae,
                    const int* src, const int* dst, const float* edge_attr, float* hg,
                    float* out, const Scratch& S) {
  constexpr int COLT = C / 16;
  if (ET) wa_k<<<1, 16, 0, st>>>(We, ae, S.wa, C);
  gemm_k128<COLT><<<NN / 16, 32 * COLT, 0, st>>>(hin, Wm, hg);
  node_prep_k<C, ET><<<(NN * 32 + 255) / 256, 256, 0, st>>>(hg, asrc, adst, S.mean_attr,
                                                            S.wa, S.s_src, S.s_dst,
                                                            S.self_l, S.segmax_u);
  if (ET) eterm_k<<<(NE * 16 + 255) / 256, 256, 0, st>>>(edge_attr, S.wa, S.eterm);
  edge_max_k<ET><<<(NE + 255) / 256, 256, 0, st>>>(src, dst, S.s_src, S.s_dst, S.eterm,
                                                   S.segmax_u);
  node_mid_k<<<(NN + 255) / 256, 256, 0, st>>>(S.segmax_u, S.self_l, S.segmaxf, S.segsum);
  edge_sum_k<ET><<<(NE + 255) / 256, 256, 0, st>>>(src, dst, S.s_src, S.s_dst, S.eterm,
                                                   S.segmaxf, S.segsum);
  out_init_k<C><<<(NN * (C / 4) + 255) / 256, 256, 0, st>>>(hg, S.self_l, S.segmaxf,
                                                            S.segsum, bias, out);
  edge_agg_k<C, ET><<<(NE * 32 + 255) / 256, 256, 0, st>>>(src, dst, S.s_src, S.s_dst,
                                                           S.eterm, S.segmaxf, S.segsum,
                                                           hg, out);
}

extern "C" void kernel_launch(void* const* d_in, const int* in_sizes, int n_in, void* d_out,
                              int out_size, void* d_ws, size_t ws_size, hipStream_t stream) {
  (void)in_sizes; (void)n_in; (void)out_size; (void)ws_size;
  const float* x = (const float*)d_in[0];
  const int* ei = (const int*)d_in[1];
  const int* src = ei;           // edge_index[0]
  const int* dst = ei + NE;      // edge_index[1]
  const float* edge_attr = (const float*)d_in[2];
  const float *W1 = (const float*)d_in[3], *a1s = (const float*)d_in[4],
              *a1d = (const float*)d_in[5], *b1 = (const float*)d_in[6];
  const float *W2 = (const float*)d_in[7], *a2s = (const float*)d_in[8],
              *a2d = (const float*)d_in[9], *b2 = (const float*)d_in[10];
  const float *We2 = (const float*)d_in[11], *ae2 = (const float*)d_in[12];
  const float *Wr = (const float*)d_in[13], *br = (const float*)d_in[14];
  const float *Wmu = (const float*)d_in[15], *amus = (const float*)d_in[16],
              *amud = (const float*)d_in[17], *bmu = (const float*)d_in[18],
              *Wemu = (const float*)d_in[19], *aemu = (const float*)d_in[20];
  const float *Wls = (const float*)d_in[21], *alss = (const float*)d_in[22],
              *alsd = (const float*)d_in[23], *bls = (const float*)d_in[24],
              *Wels = (const float*)d_in[25], *aels = (const float*)d_in[26];
  float* outf = (float*)d_out;  // [mu | logstd], each NN*64 floats

  // workspace layout (floats)
  float* W = (float*)d_ws;
  size_t o = 0;
  float* hA = W + o; o += (size_t)NN * 128;
  float* hB = W + o; o += (size_t)NN * 128;
  float* hC = W + o; o += (size_t)NN * 128;
  Scratch S;
  S.s_src = W + o; o += NN;
  S.s_dst = W + o; o += NN;
  S.self_l = W + o; o += NN;
  S.segmaxf = W + o; o += NN;
  S.segsum = W + o; o += NN;
  S.segmax_u = (unsigned*)(W + o); o += NN;
  float* cnt = W + o; o += NN;
  float* mean_attr = W + o; o += (size_t)NN * 16;  // contiguous after cnt
  S.mean_attr = mean_attr;
  S.eterm = W + o; o += NE;
  S.wa = W + o; o += 16;

  // shared: per-destination mean of edge_attr (used by conv2/mu/logstd self loops)
  zero_f32_k<<<(NN * 17 + 255) / 256, 256, 0, stream>>>(cnt, NN * 17);
  attr_accum_k<<<(NE * 16 + 255) / 256, 256, 0, stream>>>(dst, edge_attr, mean_attr, cnt);
  attr_mean_k<<<(NN * 16 + 255) / 256, 256, 0, stream>>>(mean_attr, cnt);

  // conv1: h1 = relu(GAT(x)) -> hB
  run_gat<128, false>(stream, x, W1, a1s, a1d, b1, nullptr, nullptr, src, dst, edge_attr,
                      hA, hB, S);
  relu_k<<<(NN * 128 + 255) / 256, 256, 0, stream>>>(hB, NN * 128);

  // conv2 (with edge term): out -> hC (pre-relu)
  run_gat<128, true>(stream, hB, W2, a2s, a2d, b2, We2, ae2, src, dst, edge_attr, hA, hC, S);

  // initial residual: h3 = relu(hC) + x@Wr + br   (into hB; hB free after conv2 GEMM)
  gemm_k128<8><<<NN / 16, 256, 0, stream>>>(x, Wr, hB);
  residual_k<<<(NN * 128 + 255) / 256, 256, 0, stream>>>(hC, hB, br, NN * 128);

  // mu and logstd heads (COUT=64), written straight into d_out
  run_gat<64, true>(stream, hB, Wmu, amus, amud, bmu, Wemu, aemu, src, dst, edge_attr, hA,
                    outf, S);
  run_gat<64, true>(stream, hB, Wls, alss, alsd, bls, Wels, aels, src, dst, edge_attr, hA,
                    outf + (size_t)NN * 64, S);
}